// TopLabel_Specific_ECE_LOSS_equal_width_62448824484164
// MI455X (gfx1250) — compile-verified
//
#include <hip/hip_runtime.h>
#include <hip/hip_bf16.h>
#include <stdint.h>

// ---- problem constants ----
#define NBINS            15
#define NCLASS           100
#define NSEG             (NCLASS * NBINS)      // 1500
#define ROW_BYTES        400                   // 100 f32
#define CHUNKS_PER_ROW   25                    // 400 / 16
#define WAVES_PER_BLOCK  8                     // 256 threads, wave32
#define ROWS_PER_WAVE    8
#define TILE_ROWS        (WAVES_PER_BLOCK * ROWS_PER_WAVE)   // 64
#define WAVE_BYTES       (ROWS_PER_WAVE * ROW_BYTES)         // 3200
#define TILE_BYTES       (TILE_ROWS * ROW_BYTES)             // 25600
#define CHUNKS_PER_WAVE  (ROWS_PER_WAVE * CHUNKS_PER_ROW)    // 200
#define WS_FLOATS        (3 * NSEG + NCLASS)                 // 4600
#define MAX_BLOCKS       768

// LDS byte offset of a __shared__ object: generic -> addrspace(3) cast strips
// the flat aperture, leaving the workgroup-LDS offset the async ops want.
__device__ __forceinline__ unsigned int lds_offset_of(const void* p) {
    return (unsigned int)(size_t)(__attribute__((address_space(3))) const void*)p;
}

// CDNA5 async global->LDS copy: 16B per active lane (512B per wave-op).
// Tracked by ASYNCcnt.
__device__ __forceinline__ void async_copy_b128(unsigned int lds_off,
                                                unsigned long long gbase,
                                                unsigned int goff) {
    asm volatile("global_load_async_to_lds_b128 %0, %1, %2"
                 :
                 : "v"(lds_off), "v"(goff), "s"(gbase)
                 : "memory");
}

__device__ __forceinline__ void wait_async0() {
    asm volatile("s_wait_asynccnt 0" ::: "memory");
}

__global__ void ece_zero_ws_kernel(float* __restrict__ ws, int n) {
    int i = blockIdx.x * blockDim.x + threadIdx.x;
    if (i < n) ws[i] = 0.0f;
}

__global__ __launch_bounds__(256)
void ece_main_kernel(const float* __restrict__ y_pred,
                     const int*   __restrict__ y_true,
                     float*       __restrict__ ws,
                     int N) {
    // 2 x 25600 B tile double-buffer + 12.4 KB histograms = 63.6 KB LDS
    __shared__ __align__(16) unsigned char s_tiles[2 * TILE_BYTES];
    __shared__ unsigned int s_ca[NSEG];     // packed: cnt in [15:0], acc in [31:16]
    __shared__ float        s_csum[NSEG];
    __shared__ unsigned int s_lbl[NCLASS];

    const int tid  = threadIdx.x;
    const int wave = tid >> 5;
    const int lane = tid & 31;

    for (int i = tid; i < NSEG; i += blockDim.x) { s_ca[i] = 0u; s_csum[i] = 0.0f; }
    for (int i = tid; i < NCLASS; i += blockDim.x) s_lbl[i] = 0u;
    __syncthreads();

    const unsigned long long gbase = (unsigned long long)(size_t)y_pred;
    const unsigned int ldsTileBase = lds_offset_of(&s_tiles[0]);
    const int numTiles = (N + TILE_ROWS - 1) / TILE_ROWS;

    // ---- per-wave async issue of this wave's 8 rows of a tile ----
    auto issue = [&](int tile, int buf) {
        const int r0 = tile * TILE_ROWS + wave * ROWS_PER_WAVE;
        const unsigned long long rowByte = (unsigned long long)r0 * ROW_BYTES;
        const unsigned int lbase = ldsTileBase + (unsigned)(buf * TILE_BYTES + wave * WAVE_BYTES);
        for (int c = lane; c < CHUNKS_PER_WAVE; c += 32) {
            const int row = r0 + c / CHUNKS_PER_ROW;
            if (row < N) {
                async_copy_b128(lbase + (unsigned)(c * 16), gbase,
                                (unsigned int)(rowByte + (unsigned)(c * 16)));
            }
        }
    };

    // ---- per-wave processing of this wave's 8 rows from LDS ----
    auto process = [&](int tile, int buf) {
        const int r0 = tile * TILE_ROWS + wave * ROWS_PER_WAVE;
        const unsigned char* wbase = &s_tiles[buf * TILE_BYTES + wave * WAVE_BYTES];
        for (int rr = 0; rr < ROWS_PER_WAVE; ++rr) {
            const int row = r0 + rr;
            if (row >= N) break;
            const float4* rp = reinterpret_cast<const float4*>(wbase + rr * ROW_BYTES);
            float4 x = make_float4(0.f, 0.f, 0.f, 0.f);
            float m = -3.0e38f;
            int   mi = 0x7FFFFFF;                 // losing-lane sentinel (never wins ties)
            if (lane < CHUNKS_PER_ROW) {
                x = rp[lane];
                m = x.x; mi = lane * 4;
                if (x.y > m) { m = x.y; mi = lane * 4 + 1; }
                if (x.z > m) { m = x.z; mi = lane * 4 + 2; }
                if (x.w > m) { m = x.w; mi = lane * 4 + 3; }
            }
            // wave32 allreduce: max value, first-occurrence index on ties
            #pragma unroll
            for (int off = 16; off > 0; off >>= 1) {
                float om = __shfl_xor(m, off, 32);
                int   oi = __shfl_xor(mi, off, 32);
                if (om > m || (om == m && oi < mi)) { m = om; mi = oi; }
            }
            float s = 0.0f;
            if (lane < CHUNKS_PER_ROW) {
                s = expf(x.x - m) + expf(x.y - m) + expf(x.z - m) + expf(x.w - m);
            }
            #pragma unroll
            for (int off = 16; off > 0; off >>= 1) s += __shfl_xor(s, off, 32);

            if (lane == 0) {
                const float conf  = 1.0f / s;      // softmax top-label probability
                const int   pred  = mi;
                const int   label = y_true[row];
                const unsigned int accBit = (pred == label) ? 1u : 0u;
                int b = (int)ceilf(conf * (float)NBINS) - 1;
                b = b < 0 ? 0 : (b > NBINS - 1 ? NBINS - 1 : b);
                const int seg = pred * NBINS + b;
                atomicAdd(&s_ca[seg], 1u | (accBit << 16));   // cnt + acc in one ds_add
                atomicAdd(&s_csum[seg], conf);
                atomicAdd(&s_lbl[label], 1u);
            }
        }
    };

    // ---- double-buffered async pipeline (per-wave, no block barriers) ----
    int buf = 0;
    int t = blockIdx.x;
    if (t < numTiles) issue(t, buf);
    wait_async0();
    for (; t < numTiles; t += gridDim.x) {
        const int tn = t + gridDim.x;
        if (tn < numTiles) issue(tn, buf ^ 1);
        process(t, buf);
        wait_async0();            // next buffer's async loads complete before swap
        buf ^= 1;
    }

    // ---- flush block-private histograms (sparsity-guarded global atomics) ----
    __syncthreads();
    for (int i = tid; i < NSEG; i += blockDim.x) {
        const unsigned int p = s_ca[i];
        if (p != 0u) {
            atomicAdd(&ws[i],            (float)(p & 0xFFFFu));   // cnt
            atomicAdd(&ws[NSEG + i],     s_csum[i]);              // conf sum
            atomicAdd(&ws[2 * NSEG + i], (float)(p >> 16));       // acc sum
        }
    }
    for (int i = tid; i < NCLASS; i += blockDim.x) {
        const unsigned int c = s_lbl[i];
        if (c != 0u) atomicAdd(&ws[3 * NSEG + i], (float)c);
    }
}

__global__ void ece_finalize_kernel(const float* __restrict__ ws,
                                    float* __restrict__ out) {
    __shared__ float red_e[128];
    __shared__ float red_p[128];
    const int c = threadIdx.x;
    float e = 0.0f, p = 0.0f;
    if (c < NCLASS) {
        const float* cnt  = ws + c * NBINS;
        const float* csum = ws + NSEG + c * NBINS;
        const float* asum = ws + 2 * NSEG + c * NBINS;
        float class_n = 0.0f;
        #pragma unroll
        for (int b = 0; b < NBINS; ++b) class_n += cnt[b];
        const float cn = fmaxf(class_n, 1.0f);
        float ece_c = 0.0f;
        #pragma unroll
        for (int b = 0; b < NBINS; ++b) {
            const float n = cnt[b];
            if (n > 0.0f) {
                const float sc = fmaxf(n, 1.0f);
                ece_c += fabsf(csum[b] / sc - asum[b] / sc) * (n / cn);
            }
        }
        const float present = (ws[3 * NSEG + c] > 0.0f) ? 1.0f : 0.0f;
        e = ece_c * present;
        p = present;
    }
    red_e[c] = e; red_p[c] = p;
    __syncthreads();
    for (int s = 64; s > 0; s >>= 1) {
        if (c < s) { red_e[c] += red_e[c + s]; red_p[c] += red_p[c + s]; }
        __syncthreads();
    }
    if (c == 0) out[0] = red_e[0] / fmaxf(red_p[0], 1.0f);
}

extern "C" void kernel_launch(void* const* d_in, const int* in_sizes, int n_in,
                              void* d_out, int out_size, void* d_ws, size_t ws_size,
                              hipStream_t stream) {
    const float* y_pred = (const float*)d_in[0];
    const int*   y_true = (const int*)d_in[1];
    const int N = in_sizes[1];            // 1,000,000 rows; in_sizes[0] = N*100
    float* ws  = (float*)d_ws;            // 4600 floats of scratch
    float* out = (float*)d_out;

    ece_zero_ws_kernel<<<(WS_FLOATS + 255) / 256, 256, 0, stream>>>(ws, WS_FLOATS);

    const int numTiles = (N + TILE_ROWS - 1) / TILE_ROWS;
    const int blocks = numTiles < MAX_BLOCKS ? numTiles : MAX_BLOCKS;
    ece_main_kernel<<<blocks, 256, 0, stream>>>(y_pred, y_true, ws, N);

    ece_finalize_kernel<<<1, 128, 0, stream>>>(ws, out);
}